// EdgeGatedGraphConv_86809878987275
// MI455X (gfx1250) — compile-verified
//
#include <hip/hip_runtime.h>
#include <cstdint>

#define LDSW 68   // padded row stride (floats) for 64-wide tiles in LDS

typedef __attribute__((ext_vector_type(2))) float v2f;
typedef __attribute__((ext_vector_type(8))) float v8f;

static __device__ __forceinline__ v8f wmma4(v2f a, v2f b, v8f c) {
  // D(16x16,f32) = A(16x4,f32) * B(4x16,f32) + C
  return __builtin_amdgcn_wmma_f32_16x16x4_f32(false, a, false, b, (short)0, c,
                                               false, false);
}

static __device__ __forceinline__ float xor16_sum(float v) {
  v += __shfl_xor(v, 1, 32);
  v += __shfl_xor(v, 2, 32);
  v += __shfl_xor(v, 4, 32);
  v += __shfl_xor(v, 8, 32);
  return v;
}

static __device__ __forceinline__ void atomicMaxF(float* addr, float v) {
  if (v >= 0.0f) atomicMax((int*)addr, __float_as_int(v));
  else           atomicMin((unsigned int*)addr, __float_as_uint(v));
}

// ---------------------------------------------------------------------------
// Generic OUT[rows,64] = X[rows,64] @ W[64,64]^T + bias, fp32 WMMA.
// One wave = 16 rows; 256 threads = 8 waves = 128 rows / block.
// ---------------------------------------------------------------------------
__global__ __launch_bounds__(256) void gemm64_kernel(
    const float* __restrict__ X, const float* __restrict__ W,
    const float* __restrict__ bias, float* __restrict__ out, int nrows) {
  __shared__ float lw[64 * LDSW];
  __shared__ float la[8][16 * LDSW];
  const int tid = threadIdx.x;
  for (int i = tid; i < 4096; i += 256) lw[(i >> 6) * LDSW + (i & 63)] = W[i];
  const int wave = tid >> 5, lane = tid & 31;
  const int half = lane >> 4, l16 = lane & 15;
  const int rowBase = (blockIdx.x * 8 + wave) * 16;
  float* A = la[wave];
  {
    const int r = rowBase + l16;
    const bool ok = r < nrows;
    const float* p = X + (size_t)(ok ? r : 0) * 64 + 32 * half;
#pragma unroll
    for (int j = 0; j < 8; ++j) {
      float4 v = ok ? *(const float4*)(p + 4 * j) : make_float4(0.f, 0.f, 0.f, 0.f);
      *(float4*)&A[l16 * LDSW + 32 * half + 4 * j] = v;
    }
  }
  __syncthreads();
  v8f acc[4] = {};
#pragma unroll
  for (int k0 = 0; k0 < 64; k0 += 4) {
    const int ka = k0 + 2 * half;
    v2f a; a.x = A[l16 * LDSW + ka]; a.y = A[l16 * LDSW + ka + 1];
#pragma unroll
    for (int t = 0; t < 4; ++t) {
      const int n = t * 16 + l16;
      v2f b; b.x = lw[n * LDSW + ka]; b.y = lw[n * LDSW + ka + 1];
      acc[t] = wmma4(a, b, acc[t]);
    }
  }
#pragma unroll
  for (int t = 0; t < 4; ++t) {
    const int col = t * 16 + l16;
    const float bv = bias[col];
#pragma unroll
    for (int r = 0; r < 8; ++r) {
      const int row = rowBase + r + 8 * half;
      if (row < nrows) out[(size_t)row * 64 + col] = acc[t][r] + bv;
    }
  }
}

// a_src[n] = dot(e_src[n], att_src), a_dst[n] = dot(e_dst[n], att_dst)
__global__ void att_kernel(const float* __restrict__ e_src,
                           const float* __restrict__ e_dst,
                           const float* __restrict__ att_s,
                           const float* __restrict__ att_d,
                           float* __restrict__ a_s, float* __restrict__ a_d,
                           int n) {
  int i = blockIdx.x * blockDim.x + threadIdx.x;
  if (i >= n) return;
  const float* ps = e_src + (size_t)i * 64;
  const float* pd = e_dst + (size_t)i * 64;
  float s = 0.f, d = 0.f;
#pragma unroll 8
  for (int j = 0; j < 64; ++j) { s += ps[j] * att_s[j]; d += pd[j] * att_d[j]; }
  a_s[i] = s; a_d[i] = d;
}

__global__ void init_kernel(float* mx_s, float* mx_d, float* den_s, float* den_d,
                            float* ssh, float* ss, int n) {
  int i = blockIdx.x * blockDim.x + threadIdx.x;
  if (i < n) {
    mx_s[i] = -INFINITY; mx_d[i] = -INFINITY;
    den_s[i] = 0.f; den_d[i] = 0.f;
  }
  if (i < n * 16) {
    float4 z = make_float4(0.f, 0.f, 0.f, 0.f);
    ((float4*)ssh)[i] = z;
    ((float4*)ss)[i] = z;
  }
}

__global__ void edge_max_kernel(const int* __restrict__ src, const int* __restrict__ dst,
                                const float* __restrict__ a_s, const float* __restrict__ a_d,
                                float* mx_s, float* mx_d, int E) {
  int e = blockIdx.x * blockDim.x + threadIdx.x;
  if (e >= E) return;
  int s = src[e], d = dst[e];
  atomicMaxF(&mx_s[d], a_s[s]);
  atomicMaxF(&mx_d[d], a_d[s]);
}

__global__ void edge_den_kernel(const int* __restrict__ src, const int* __restrict__ dst,
                                const float* __restrict__ a_s, const float* __restrict__ a_d,
                                const float* __restrict__ mx_s, const float* __restrict__ mx_d,
                                float* den_s, float* den_d, int E) {
  int e = blockIdx.x * blockDim.x + threadIdx.x;
  if (e >= E) return;
  int s = src[e], d = dst[e];
  atomicAdd(&den_s[d], expf(a_s[s] - mx_s[d]));
  atomicAdd(&den_d[d], expf(a_d[s] - mx_d[d]));
}

// ---------------------------------------------------------------------------
// Fused edge pipeline. One wave = 16 edges; 128 threads = 4 waves = 64 edges.
//   m = att_s*e_src[src] + att_d*e_dst[src] + edge_gate(edge_feats)    (WMMA)
//   sigma = sigmoid(m); atomic segment sums of Bh[src]*sigma and sigma
//   y = edge_feats + silu(LN(m_gate(m)))                               (WMMA)
// ---------------------------------------------------------------------------
__global__ __launch_bounds__(128) void edge_main_kernel(
    const float* __restrict__ edge_feats, const int* __restrict__ src,
    const int* __restrict__ dst,
    const float* __restrict__ eg_w, const float* __restrict__ eg_b,
    const float* __restrict__ mg_w, const float* __restrict__ mg_b,
    const float* __restrict__ e_src_g, const float* __restrict__ e_dst_g,
    const float* __restrict__ Bh,
    const float* __restrict__ a_s, const float* __restrict__ a_d,
    const float* __restrict__ mx_s, const float* __restrict__ mx_d,
    const float* __restrict__ den_s, const float* __restrict__ den_d,
    const float* __restrict__ ln_g, const float* __restrict__ ln_b,
    float* __restrict__ sum_sigma_h, float* __restrict__ sum_sigma,
    float* __restrict__ y_out, int E) {
  __shared__ float lw1[64 * LDSW];          // edge_gate_w
  __shared__ float lw2[64 * LDSW];          // m_gate_w
  __shared__ float m_t[4][16 * LDSW];       // gathered part of m -> m -> y
  __shared__ float x_t[4][16 * LDSW];       // edge_feats tile
  __shared__ float bh_t[4][16 * LDSW];      // Bh[src] tile
  __shared__ int dR[4][16];

  const int tid = threadIdx.x;
  for (int i = tid; i < 4096; i += 128) {
    lw1[(i >> 6) * LDSW + (i & 63)] = eg_w[i];
    lw2[(i >> 6) * LDSW + (i & 63)] = mg_w[i];
  }
  const int wave = tid >> 5, lane = tid & 31;
  const int half = lane >> 4, l16 = lane & 15;
  const int ebase = (blockIdx.x * 4 + wave) * 16;
  const int nvalid = min(16, E - ebase);
  const int er = ebase + l16;
  const int e = (er < E) ? er : (E - 1);
  const int si = src[e], di = dst[e];
  if (half == 0) dR[wave][l16] = di;
  const float as = expf(a_s[si] - mx_s[di]) / den_s[di];
  const float ad = expf(a_d[si] - mx_d[di]) / den_d[di];
  {
    const size_t so = (size_t)si * 64 + 32 * half;
    const size_t eo = (size_t)e * 64 + 32 * half;
    float* mrow = &m_t[wave][l16 * LDSW + 32 * half];
    float* xrow = &x_t[wave][l16 * LDSW + 32 * half];
    float* brow = &bh_t[wave][l16 * LDSW + 32 * half];
#pragma unroll
    for (int j = 0; j < 8; ++j) {
      float4 vs = *(const float4*)(e_src_g + so + 4 * j);
      float4 vd = *(const float4*)(e_dst_g + so + 4 * j);
      float4 vb = *(const float4*)(Bh + so + 4 * j);
      float4 ve = *(const float4*)(edge_feats + eo + 4 * j);
      float4 vm;
      vm.x = as * vs.x + ad * vd.x;
      vm.y = as * vs.y + ad * vd.y;
      vm.z = as * vs.z + ad * vd.z;
      vm.w = as * vs.w + ad * vd.w;
      *(float4*)(mrow + 4 * j) = vm;
      *(float4*)(xrow + 4 * j) = ve;
      *(float4*)(brow + 4 * j) = vb;
    }
  }
  __syncthreads();  // weights visible to all waves

  // ---- GEMM1: edge_gate(edge_feats) ----
  v8f acc[4] = {};
  {
    const float* A = x_t[wave];
#pragma unroll
    for (int k0 = 0; k0 < 64; k0 += 4) {
      const int ka = k0 + 2 * half;
      v2f a; a.x = A[l16 * LDSW + ka]; a.y = A[l16 * LDSW + ka + 1];
#pragma unroll
      for (int t = 0; t < 4; ++t) {
        const int n = t * 16 + l16;
        v2f b; b.x = lw1[n * LDSW + ka]; b.y = lw1[n * LDSW + ka + 1];
        acc[t] = wmma4(a, b, acc[t]);
      }
    }
  }
  // ---- m, sigma, segment-sum atomics; write m back to LDS (A for GEMM2) ----
#pragma unroll
  for (int t = 0; t < 4; ++t) {
    const int col = t * 16 + l16;
    const float bv = eg_b[col];
#pragma unroll
    for (int r = 0; r < 8; ++r) {
      const int row = r + 8 * half;
      const float mval = acc[t][r] + bv + m_t[wave][row * LDSW + col];
      const float sg = 1.0f / (1.0f + expf(-mval));
      if (row < nvalid) {
        const int dn = dR[wave][row];
        atomicAdd(&sum_sigma_h[(size_t)dn * 64 + col],
                  bh_t[wave][row * LDSW + col] * sg);
        atomicAdd(&sum_sigma[(size_t)dn * 64 + col], sg);
      }
      m_t[wave][row * LDSW + col] = mval;
    }
  }
  // ---- GEMM2: m_gate(m) ----
  v8f acc2[4] = {};
  {
    const float* A = m_t[wave];
#pragma unroll
    for (int k0 = 0; k0 < 64; k0 += 4) {
      const int ka = k0 + 2 * half;
      v2f a; a.x = A[l16 * LDSW + ka]; a.y = A[l16 * LDSW + ka + 1];
#pragma unroll
      for (int t = 0; t < 4; ++t) {
        const int n = t * 16 + l16;
        v2f b; b.x = lw2[n * LDSW + ka]; b.y = lw2[n * LDSW + ka + 1];
        acc2[t] = wmma4(a, b, acc2[t]);
      }
    }
  }
  // ---- LayerNorm + SiLU + residual ----
  float vals[4][8], gc[4], bc[4];
#pragma unroll
  for (int t = 0; t < 4; ++t) {
    const int col = t * 16 + l16;
    const float bv = mg_b[col];
    gc[t] = ln_g[col]; bc[t] = ln_b[col];
#pragma unroll
    for (int r = 0; r < 8; ++r) vals[t][r] = acc2[t][r] + bv;
  }
  float mu[8], rstd[8];
#pragma unroll
  for (int r = 0; r < 8; ++r) {
    float s = 0.f, q = 0.f;
#pragma unroll
    for (int t = 0; t < 4; ++t) { float v = vals[t][r]; s += v; q += v * v; }
    s = xor16_sum(s); q = xor16_sum(q);
    const float m = s * (1.0f / 64.0f);
    mu[r] = m;
    rstd[r] = rsqrtf(q * (1.0f / 64.0f) - m * m + 1e-5f);
  }
#pragma unroll
  for (int t = 0; t < 4; ++t) {
    const int col = t * 16 + l16;
#pragma unroll
    for (int r = 0; r < 8; ++r) {
      const int row = r + 8 * half;
      const float ln = (vals[t][r] - mu[r]) * rstd[r] * gc[t] + bc[t];
      const float si2 = ln / (1.0f + expf(-ln));
      m_t[wave][row * LDSW + col] = x_t[wave][row * LDSW + col] + si2;
    }
  }
  // ---- coalesced store of y ----
  if (l16 < nvalid) {
    const size_t eo = (size_t)(ebase + l16) * 64 + 32 * half;
    const float* mrow = &m_t[wave][l16 * LDSW + 32 * half];
#pragma unroll
    for (int j = 0; j < 8; ++j)
      *(float4*)(y_out + eo + 4 * j) = *(const float4*)(mrow + 4 * j);
  }
}

// ---------------------------------------------------------------------------
// Node finalize: x = node_feats + silu(LN(x_gate(xs + ssh/(ss+1e-6))))
// ---------------------------------------------------------------------------
__global__ __launch_bounds__(256) void node_final_kernel(
    const float* __restrict__ node_feats, const float* __restrict__ xs,
    const float* __restrict__ ssh, const float* __restrict__ ss,
    const float* __restrict__ xg_w, const float* __restrict__ xg_b,
    const float* __restrict__ ln_g, const float* __restrict__ ln_b,
    float* __restrict__ x_out, int n) {
  __shared__ float lw[64 * LDSW];
  __shared__ float la[8][16 * LDSW];
  const int tid = threadIdx.x;
  for (int i = tid; i < 4096; i += 256) lw[(i >> 6) * LDSW + (i & 63)] = xg_w[i];
  const int wave = tid >> 5, lane = tid & 31;
  const int half = lane >> 4, l16 = lane & 15;
  const int rowBase = (blockIdx.x * 8 + wave) * 16;
  float* A = la[wave];
  {
    const int r = rowBase + l16;
    const bool ok = r < n;
    const size_t off = (size_t)(ok ? r : 0) * 64 + 32 * half;
#pragma unroll
    for (int j = 0; j < 8; ++j) {
      float4 vx = *(const float4*)(xs + off + 4 * j);
      float4 vh = *(const float4*)(ssh + off + 4 * j);
      float4 vs = *(const float4*)(ss + off + 4 * j);
      float4 a;
      a.x = ok ? vx.x + vh.x / (vs.x + 1e-6f) : 0.f;
      a.y = ok ? vx.y + vh.y / (vs.y + 1e-6f) : 0.f;
      a.z = ok ? vx.z + vh.z / (vs.z + 1e-6f) : 0.f;
      a.w = ok ? vx.w + vh.w / (vs.w + 1e-6f) : 0.f;
      *(float4*)&A[l16 * LDSW + 32 * half + 4 * j] = a;
    }
  }
  __syncthreads();
  v8f acc[4] = {};
#pragma unroll
  for (int k0 = 0; k0 < 64; k0 += 4) {
    const int ka = k0 + 2 * half;
    v2f a; a.x = A[l16 * LDSW + ka]; a.y = A[l16 * LDSW + ka + 1];
#pragma unroll
    for (int t = 0; t < 4; ++t) {
      const int nn = t * 16 + l16;
      v2f b; b.x = lw[nn * LDSW + ka]; b.y = lw[nn * LDSW + ka + 1];
      acc[t] = wmma4(a, b, acc[t]);
    }
  }
  float vals[4][8], gc[4], bc[4];
#pragma unroll
  for (int t = 0; t < 4; ++t) {
    const int col = t * 16 + l16;
    const float bv = xg_b[col];
    gc[t] = ln_g[col]; bc[t] = ln_b[col];
#pragma unroll
    for (int r = 0; r < 8; ++r) vals[t][r] = acc[t][r] + bv;
  }
  float mu[8], rstd[8];
#pragma unroll
  for (int r = 0; r < 8; ++r) {
    float s = 0.f, q = 0.f;
#pragma unroll
    for (int t = 0; t < 4; ++t) { float v = vals[t][r]; s += v; q += v * v; }
    s = xor16_sum(s); q = xor16_sum(q);
    const float m = s * (1.0f / 64.0f);
    mu[r] = m;
    rstd[r] = rsqrtf(q * (1.0f / 64.0f) - m * m + 1e-5f);
  }
#pragma unroll
  for (int t = 0; t < 4; ++t) {
    const int col = t * 16 + l16;
#pragma unroll
    for (int r = 0; r < 8; ++r) {
      const int row = rowBase + r + 8 * half;
      if (row < n) {
        const float ln = (vals[t][r] - mu[r]) * rstd[r] * gc[t] + bc[t];
        const float si = ln / (1.0f + expf(-ln));
        x_out[(size_t)row * 64 + col] = node_feats[(size_t)row * 64 + col] + si;
      }
    }
  }
}

// ---------------------------------------------------------------------------
extern "C" void kernel_launch(void* const* d_in, const int* in_sizes, int n_in,
                              void* d_out, int out_size, void* d_ws, size_t ws_size,
                              hipStream_t stream) {
  (void)n_in; (void)out_size; (void)ws_size;
  const float* node_feats   = (const float*)d_in[0];
  const float* edge_feats   = (const float*)d_in[1];
  const int*   src          = (const int*)d_in[2];
  const int*   dst          = (const int*)d_in[3];
  const float* src_gate_w   = (const float*)d_in[4];
  const float* src_gate_b   = (const float*)d_in[5];
  const float* dst_gate_w   = (const float*)d_in[6];
  const float* dst_gate_b   = (const float*)d_in[7];
  const float* edge_gate_w  = (const float*)d_in[8];
  const float* edge_gate_b  = (const float*)d_in[9];
  const float* x_gate_w     = (const float*)d_in[10];
  const float* x_gate_b     = (const float*)d_in[11];
  const float* m_gate_w     = (const float*)d_in[12];
  const float* m_gate_b     = (const float*)d_in[13];
  const float* src_update_w = (const float*)d_in[14];
  const float* src_update_b = (const float*)d_in[15];
  const float* dst_update_w = (const float*)d_in[16];
  const float* dst_update_b = (const float*)d_in[17];
  const float* att_src      = (const float*)d_in[18];
  const float* att_dst      = (const float*)d_in[19];
  const float* ln_n_g       = (const float*)d_in[20];
  const float* ln_n_b       = (const float*)d_in[21];
  const float* ln_e_g       = (const float*)d_in[22];
  const float* ln_e_b       = (const float*)d_in[23];

  const int N = in_sizes[0] / 64;
  const int E = in_sizes[2];

  float* ws    = (float*)d_ws;
  float* e_src = ws; ws += (size_t)N * 64;
  float* e_dst = ws; ws += (size_t)N * 64;
  float* Bh    = ws; ws += (size_t)N * 64;
  float* xs    = ws; ws += (size_t)N * 64;
  float* ssh   = ws; ws += (size_t)N * 64;   // sum_sigma_h
  float* ssum  = ws; ws += (size_t)N * 64;   // sum_sigma
  float* a_s   = ws; ws += N;
  float* a_d   = ws; ws += N;
  float* mxs   = ws; ws += N;
  float* mxd   = ws; ws += N;
  float* dens  = ws; ws += N;
  float* dend  = ws; ws += N;

  const int gGemm = (N + 127) / 128;
  gemm64_kernel<<<gGemm, 256, 0, stream>>>(node_feats, src_gate_w, src_gate_b, e_src, N);
  gemm64_kernel<<<gGemm, 256, 0, stream>>>(node_feats, dst_gate_w, dst_gate_b, e_dst, N);
  gemm64_kernel<<<gGemm, 256, 0, stream>>>(node_feats, dst_update_w, dst_update_b, Bh, N);
  gemm64_kernel<<<gGemm, 256, 0, stream>>>(node_feats, src_update_w, src_update_b, xs, N);

  att_kernel<<<(N + 255) / 256, 256, 0, stream>>>(e_src, e_dst, att_src, att_dst, a_s, a_d, N);
  init_kernel<<<(N * 16 + 255) / 256, 256, 0, stream>>>(mxs, mxd, dens, dend, ssh, ssum, N);
  edge_max_kernel<<<(E + 255) / 256, 256, 0, stream>>>(src, dst, a_s, a_d, mxs, mxd, E);
  edge_den_kernel<<<(E + 255) / 256, 256, 0, stream>>>(src, dst, a_s, a_d, mxs, mxd, dens, dend, E);

  float* y_out = (float*)d_out + (size_t)N * 64;
  edge_main_kernel<<<(E + 63) / 64, 128, 0, stream>>>(
      edge_feats, src, dst, edge_gate_w, edge_gate_b, m_gate_w, m_gate_b,
      e_src, e_dst, Bh, a_s, a_d, mxs, mxd, dens, dend, ln_e_g, ln_e_b,
      ssh, ssum, y_out, E);

  node_final_kernel<<<gGemm, 256, 0, stream>>>(
      node_feats, xs, ssh, ssum, x_gate_w, x_gate_b, ln_n_g, ln_n_b,
      (float*)d_out, N);
}